// DWT_block_72026601554040
// MI455X (gfx1250) — compile-verified
//
#include <hip/hip_runtime.h>
#include <math.h>

typedef __attribute__((ext_vector_type(2))) float v2f;
typedef __attribute__((ext_vector_type(8))) float v8f;

#define HW   511
#define NPIX (511 * 511)
#define HF   516
#define NHF  (516 * 516)

// normalized 5-tap gaussian, sigma=0.5
__device__ __constant__ float GK[5] = {
    2.6386448e-4f, 1.0645080e-1f, 7.8657073e-1f, 1.0645080e-1f, 2.6386448e-4f};

// ---------------------------------------------------------------------------
// Pack weights (N,K) -> K-major pair layout: wp[(k/2)*2N + 2n + (k&1)].
// Consumer lanes then fetch a v2f (k,k+1) per channel with one coalesced b64.
// ---------------------------------------------------------------------------
__global__ void k_pack_w(const float* __restrict__ w, float* __restrict__ wp,
                         int K, int N) {
  const int idx = blockIdx.x * blockDim.x + threadIdx.x;
  if (idx >= K * N) return;
  const int p = idx / (2 * N);
  const int r = idx % (2 * N);
  const int n = r >> 1, o = r & 1;
  wp[idx] = w[n * K + 2 * p + o];
}

// ---------------------------------------------------------------------------
// Kernel 1: fused 1x1 conv chain 204 -> 16 (WMMA) -> 8 -> 1 (VALU), relu each.
// ---------------------------------------------------------------------------
__global__ __launch_bounds__(256) void k_conv1x1_chain(
    const float* __restrict__ x, const float* __restrict__ wp1,
    const float* __restrict__ b1, const float* __restrict__ w2,
    const float* __restrict__ b2, const float* __restrict__ w3,
    const float* __restrict__ b3, float* __restrict__ tmp12) {
  __shared__ float lds_f1[256 * 16];
  const int lane = threadIdx.x & 31;
  const int wave = threadIdx.x >> 5;
  const int half = lane >> 4;  // 0: K+{0,1}, 1: K+{2,3} per ISA A layout
  const int l16  = lane & 15;
  const int blockPix = blockIdx.x * 256;
  const int wavePix  = blockPix + wave * 32;

  int pA0 = wavePix + l16;        // M-tile 0 pixels
  int pA1 = wavePix + 16 + l16;   // M-tile 1 pixels
  if (pA0 >= NPIX) pA0 = NPIX - 1;
  if (pA1 >= NPIX) pA1 = NPIX - 1;
  const int n = l16;  // output channel (B column)

  v8f acc0 = {};
  v8f acc1 = {};
  for (int k0 = 0; k0 < 204; k0 += 4) {
    const int ka = k0 + (half << 1);
    if (k0 + 8 < 204) {
      // WGP-scope prefetch (locality 3) of the K-step-after-next planes:
      // pull into all cache levels, consumed by this same wave shortly.
      __builtin_prefetch(&x[(size_t)(ka + 8) * NPIX + pA0], 0, 3);
      __builtin_prefetch(&x[(size_t)(ka + 8) * NPIX + pA1], 0, 3);
    }
    v2f a0, a1;
    a0.x = x[(size_t)ka * NPIX + pA0];
    a0.y = x[(size_t)(ka + 1) * NPIX + pA0];
    a1.x = x[(size_t)ka * NPIX + pA1];
    a1.y = x[(size_t)(ka + 1) * NPIX + pA1];
    const v2f bb = *(const v2f*)(wp1 + ((ka >> 1) << 5) + (n << 1));
    acc0 = __builtin_amdgcn_wmma_f32_16x16x4_f32(false, a0, false, bb, (short)0,
                                                 acc0, false, false);
    acc1 = __builtin_amdgcn_wmma_f32_16x16x4_f32(false, a1, false, bb, (short)0,
                                                 acc1, false, false);
  }

  // bias + relu, scatter to LDS: lds_f1[local_pixel][chan]
  const float bias = b1[n];
#pragma unroll
  for (int v = 0; v < 8; ++v) {
    const int m = v + half * 8;
    float f0 = acc0[v] + bias;
    float f1 = acc1[v] + bias;
    f0 = f0 > 0.f ? f0 : 0.f;
    f1 = f1 > 0.f ? f1 : 0.f;
    lds_f1[(wave * 32 + m) * 16 + n]      = f0;
    lds_f1[(wave * 32 + 16 + m) * 16 + n] = f1;
  }
  __syncthreads();

  // per-thread tail: 16 -> 8 -> 1
  const int pix = blockPix + threadIdx.x;
  float f1r[16];
#pragma unroll
  for (int c = 0; c < 16; ++c) f1r[c] = lds_f1[threadIdx.x * 16 + c];
  float acc3 = b3[0];
#pragma unroll
  for (int o = 0; o < 8; ++o) {
    float a = b2[o];
#pragma unroll
    for (int c = 0; c < 16; ++c) a += w2[o * 16 + c] * f1r[c];
    a = a > 0.f ? a : 0.f;
    acc3 += w3[o] * a;
  }
  acc3 = acc3 > 0.f ? acc3 : 0.f;
  if (pix < NPIX) tmp12[pix] = acc3;
}

// ---------------------------------------------------------------------------
// Kernel 2: Haar DWT(zero-pad 512) -> (cH,cV,cD) -> w_hi 1x1 conv (pad=1) ->
// relu -> pixel_shuffle2, fused. lf branch dropped exactly (alpha==1).
// ---------------------------------------------------------------------------
__global__ void k_hf(const float* __restrict__ tmp12,
                     const float* __restrict__ w_hi,
                     const float* __restrict__ b_hi, float* __restrict__ hf) {
  const int idx = blockIdx.x * blockDim.x + threadIdx.x;
  if (idx >= 258 * 258) return;
  const int i = idx / 258, j = idx % 258;
  float cH = 0.f, cV = 0.f, cD = 0.f;
  const bool interior = (i >= 1 && i <= 256 && j >= 1 && j <= 256);
  if (interior) {
    const int y0 = 2 * (i - 1), x0 = 2 * (j - 1);
    const float a = tmp12[y0 * HW + x0];
    const float b = (x0 + 1 < HW) ? tmp12[y0 * HW + x0 + 1] : 0.f;
    const float c = (y0 + 1 < HW) ? tmp12[(y0 + 1) * HW + x0] : 0.f;
    const float d =
        (y0 + 1 < HW && x0 + 1 < HW) ? tmp12[(y0 + 1) * HW + x0 + 1] : 0.f;
    cH = 0.5f * (a + b - c - d);
    cV = 0.5f * (a - b + c - d);
    cD = 0.5f * (a - b - c + d);
  }
#pragma unroll
  for (int o = 0; o < 4; ++o) {
    float v = b_hi[o];
    if (interior)
      v += w_hi[o * 3 + 0] * cH + w_hi[o * 3 + 1] * cV + w_hi[o * 3 + 2] * cD;
    v = v > 0.f ? v : 0.f;
    hf[(2 * i + (o >> 1)) * HF + (2 * j + (o & 1))] = v;
  }
}

__device__ __forceinline__ int refl(int p, int n) {
  if (p < 0) p = -p;
  if (p >= n) p = 2 * n - 2 - p;
  return p;
}

__global__ void k_blur_h(const float* __restrict__ in, float* __restrict__ out) {
  const int idx = blockIdx.x * blockDim.x + threadIdx.x;
  if (idx >= NHF) return;
  const int i = idx / HF, j = idx % HF;
  float s = 0.f;
#pragma unroll
  for (int k = 0; k < 5; ++k) s += GK[k] * in[i * HF + refl(j - 2 + k, HF)];
  out[idx] = s;
}

__global__ void k_blur_v(const float* __restrict__ in, float* __restrict__ out) {
  const int idx = blockIdx.x * blockDim.x + threadIdx.x;
  if (idx >= NHF) return;
  const int i = idx / HF, j = idx % HF;
  float s = 0.f;
#pragma unroll
  for (int k = 0; k < 5; ++k) s += GK[k] * in[refl(i - 2 + k, HF) * HF + j];
  out[idx] = s;
}

// deterministic two-stage reduction for channel attention (mean + max)
__global__ __launch_bounds__(256) void k_reduce(const float* __restrict__ hfb,
                                                float* __restrict__ psum,
                                                float* __restrict__ pmax) {
  __shared__ float ss[256], sm[256];
  const int tid = threadIdx.x;
  const int idx = blockIdx.x * 256 + tid;
  const float v = (idx < NHF) ? hfb[idx] : 0.f;
  ss[tid] = v;
  sm[tid] = (idx < NHF) ? v : -3.4e38f;
  __syncthreads();
  for (int s = 128; s > 0; s >>= 1) {
    if (tid < s) {
      ss[tid] += ss[tid + s];
      sm[tid] = fmaxf(sm[tid], sm[tid + s]);
    }
    __syncthreads();
  }
  if (tid == 0) {
    psum[blockIdx.x] = ss[0];
    pmax[blockIdx.x] = sm[0];
  }
}

__global__ __launch_bounds__(256) void k_scale(
    const float* __restrict__ psum, const float* __restrict__ pmax, int nblk,
    const float* __restrict__ w1, const float* __restrict__ w2,
    float* __restrict__ scale) {
  __shared__ float ss[256], sm[256];
  const int tid = threadIdx.x;
  float s = 0.f, m = -3.4e38f;
  for (int i = tid; i < nblk; i += 256) {
    s += psum[i];
    m = fmaxf(m, pmax[i]);
  }
  ss[tid] = s;
  sm[tid] = m;
  __syncthreads();
  for (int st = 128; st > 0; st >>= 1) {
    if (tid < st) {
      ss[tid] += ss[tid + st];
      sm[tid] = fmaxf(sm[tid], sm[tid + st]);
    }
    __syncthreads();
  }
  if (tid == 0) {
    const float mean = ss[0] / (float)NHF;
    const float mx = sm[0];
    const float a1 = w1[0] * mean, a2 = w1[0] * mx;
    const float z = w2[0] * (a1 > 0.f ? a1 : 0.f) + w2[0] * (a2 > 0.f ? a2 : 0.f);
    scale[0] = 1.f / (1.f + expf(-z));
  }
}

// 6x6 conv over (scale*hf, zeros) + bias + relu, + residual tmp12 -> f12
__global__ void k_cat(const float* __restrict__ hfb,
                      const float* __restrict__ tmp12,
                      const float* __restrict__ w_cat,
                      const float* __restrict__ b_cat,
                      const float* __restrict__ scale, float* __restrict__ f12) {
  const int idx = blockIdx.x * blockDim.x + threadIdx.x;
  if (idx >= NPIX) return;
  const int y = idx / HW, x = idx % HW;
  const float s = scale[0];
  float acc = 0.f;
#pragma unroll
  for (int u = 0; u < 6; ++u)
#pragma unroll
    for (int v = 0; v < 6; ++v)
      acc += w_cat[u * 6 + v] * hfb[(y + u) * HF + (x + v)];
  float cat = b_cat[0] + s * acc;
  cat = cat > 0.f ? cat : 0.f;
  f12[idx] = tmp12[idx] + cat;
}

// 3x3 conv 1 -> 32, pad=1, relu
__global__ void k_c9a(const float* __restrict__ f12, const float* __restrict__ w,
                      const float* __restrict__ b, float* __restrict__ h) {
  const int idx = blockIdx.x * blockDim.x + threadIdx.x;
  if (idx >= NPIX) return;
  const int y = idx / HW, x = idx % HW;
  float tap[9];
#pragma unroll
  for (int u = 0; u < 3; ++u)
#pragma unroll
    for (int v = 0; v < 3; ++v) {
      const int yy = y + u - 1, xx = x + v - 1;
      tap[u * 3 + v] = (yy >= 0 && yy < HW && xx >= 0 && xx < HW)
                           ? f12[yy * HW + xx]
                           : 0.f;
    }
  for (int o = 0; o < 32; ++o) {
    float acc = b[o];
#pragma unroll
    for (int t = 0; t < 9; ++t) acc += w[o * 9 + t] * tap[t];
    h[(size_t)o * NPIX + idx] = acc > 0.f ? acc : 0.f;
  }
}

// ---------------------------------------------------------------------------
// Kernel 3: 3x3 conv 32 -> 64 as WMMA GEMM. M = pixels, K = 288, N = 64.
// Interior tiles stage the 32x18x18 halo via async global->LDS DMA; boundary
// tiles fall back to the checked VGPR path. Weights come from the packed
// K-major pair layout (coalesced b64 per fragment).
// ---------------------------------------------------------------------------
__global__ __launch_bounds__(256) void k_c9b_wmma(
    const float* __restrict__ h, const float* __restrict__ wp9,
    const float* __restrict__ b, float* __restrict__ out) {
  __shared__ float hl[32 * 18 * 18];
  const int tid = threadIdx.x;
  const int lane = tid & 31, wave = tid >> 5;
  const int half = lane >> 4, l16 = lane & 15;
  const int tx = blockIdx.x & 31, ty = blockIdx.x >> 5;
  const int gx0 = tx * 16, gy0 = ty * 16;

  const bool interior = (tx >= 1 && tx <= 30 && ty >= 1 && ty <= 30);
  if (interior) {
    // CDNA5 async copy: global -> LDS, tracked by ASYNCcnt.
    for (int idx = tid; idx < 32 * 18 * 18; idx += 256) {
      const int c = idx / 324, r = idx % 324;
      const int yy = r / 18, xx = r % 18;
      const unsigned long long ga = (unsigned long long)(
          h + ((size_t)c * NPIX + (size_t)(gy0 + yy - 1) * HW + (gx0 + xx - 1)));
      const unsigned int la = (unsigned int)(uintptr_t)&hl[idx];
      asm volatile("global_load_async_to_lds_b32 %0, %1, off"
                   :
                   : "v"(la), "v"(ga)
                   : "memory");
    }
    asm volatile("s_wait_asynccnt 0x0" ::: "memory");
  } else {
    for (int idx = tid; idx < 32 * 18 * 18; idx += 256) {
      const int c = idx / 324, r = idx % 324;
      const int yy = r / 18, xx = r % 18;
      const int gy = gy0 + yy - 1, gx = gx0 + xx - 1;
      float v = 0.f;
      if (gy >= 0 && gy < HW && gx >= 0 && gx < HW)
        v = h[(size_t)c * NPIX + gy * HW + gx];
      hl[idx] = v;
    }
  }
  __syncthreads();

  v8f acc[2][4] = {};
  const int py0 = wave * 2;      // row of M-tile 0 within the 16x16 tile
  const int py1 = wave * 2 + 1;  // row of M-tile 1
  const int px = l16;            // pixel column for A lanes

  for (int k0 = 0; k0 < 288; k0 += 4) {
    const int ka = k0 + (half << 1);
    const int c0 = ka / 9, r0 = ka % 9;
    const int u0 = r0 / 3, v0 = r0 % 3;
    const int kb = ka + 1;
    const int c1 = kb / 9, r1 = kb % 9;
    const int u1 = r1 / 3, v1 = r1 % 3;
    v2f a0, a1;
    a0.x = hl[(c0 * 18 + py0 + u0) * 18 + px + v0];
    a0.y = hl[(c1 * 18 + py0 + u1) * 18 + px + v1];
    a1.x = hl[(c0 * 18 + py1 + u0) * 18 + px + v0];
    a1.y = hl[(c1 * 18 + py1 + u1) * 18 + px + v1];
#pragma unroll
    for (int nt = 0; nt < 4; ++nt) {
      const int n = nt * 16 + l16;
      const v2f bb = *(const v2f*)(wp9 + ((ka >> 1) << 7) + (n << 1));
      acc[0][nt] = __builtin_amdgcn_wmma_f32_16x16x4_f32(
          false, a0, false, bb, (short)0, acc[0][nt], false, false);
      acc[1][nt] = __builtin_amdgcn_wmma_f32_16x16x4_f32(
          false, a1, false, bb, (short)0, acc[1][nt], false, false);
    }
  }

  // writeback: lane holds channel n = nt*16 + l16, pixel M = v + half*8
#pragma unroll
  for (int mt = 0; mt < 2; ++mt) {
    const int gy = gy0 + wave * 2 + mt;
#pragma unroll
    for (int v = 0; v < 8; ++v) {
      const int gx = gx0 + v + half * 8;
      if (gy < HW && gx < HW) {
#pragma unroll
        for (int nt = 0; nt < 4; ++nt) {
          const int ch = nt * 16 + l16;
          float val = acc[mt][nt][v] + b[ch];
          out[(size_t)ch * NPIX + gy * HW + gx] = val > 0.f ? val : 0.f;
        }
      }
    }
  }
}

// ---------------------------------------------------------------------------
extern "C" void kernel_launch(void* const* d_in, const int* in_sizes, int n_in,
                              void* d_out, int out_size, void* d_ws,
                              size_t ws_size, hipStream_t stream) {
  const float* x     = (const float*)d_in[0];
  const float* w_x1  = (const float*)d_in[1];
  const float* b_x1  = (const float*)d_in[2];
  const float* w_x2  = (const float*)d_in[3];
  const float* b_x2  = (const float*)d_in[4];
  const float* w_x3  = (const float*)d_in[5];
  const float* b_x3  = (const float*)d_in[6];
  const float* w_hi  = (const float*)d_in[7];
  const float* b_hi  = (const float*)d_in[8];
  // d_in[9..14]: w_lo,b_lo,..,ca2_* — lf branch is multiplied by (1-alpha)=0
  const float* ca1w1 = (const float*)d_in[11];
  const float* ca1w2 = (const float*)d_in[12];
  const float* w_cat = (const float*)d_in[15];
  const float* b_cat = (const float*)d_in[16];
  const float* w_c9a = (const float*)d_in[17];
  const float* b_c9a = (const float*)d_in[18];
  const float* w_c9b = (const float*)d_in[19];
  const float* b_c9b = (const float*)d_in[20];
  float* out = (float*)d_out;

  float* W     = (float*)d_ws;
  float* wp1   = W;              // 204*16  = 3264  (8B aligned)
  float* wp9   = wp1 + 3264;     // 288*64  = 18432 (8B aligned)
  float* tmp12 = wp9 + 18432;
  float* f12   = tmp12 + 261122; // NPIX padded to even
  float* hf    = f12 + 261122;
  float* hft   = hf + (size_t)NHF;
  float* hfb   = hft + (size_t)NHF;
  float* psum  = hfb + (size_t)NHF;
  float* pmax  = psum + 2048;
  float* scale = pmax + 2048;
  float* hbuf  = scale + 16;

  const int nPixBlk  = (NPIX + 255) / 256;      // 1021
  const int nHfBlk   = (258 * 258 + 255) / 256;
  const int nBlurBlk = (NHF + 255) / 256;       // 1041

  k_pack_w<<<(204 * 16 + 255) / 256, 256, 0, stream>>>(w_x1, wp1, 204, 16);
  k_pack_w<<<(288 * 64 + 255) / 256, 256, 0, stream>>>(w_c9b, wp9, 288, 64);
  k_conv1x1_chain<<<nPixBlk, 256, 0, stream>>>(x, wp1, b_x1, w_x2, b_x2, w_x3,
                                               b_x3, tmp12);
  k_hf<<<nHfBlk, 256, 0, stream>>>(tmp12, w_hi, b_hi, hf);
  k_blur_h<<<nBlurBlk, 256, 0, stream>>>(hf, hft);
  k_blur_v<<<nBlurBlk, 256, 0, stream>>>(hft, hfb);
  k_reduce<<<nBlurBlk, 256, 0, stream>>>(hfb, psum, pmax);
  k_scale<<<1, 256, 0, stream>>>(psum, pmax, nBlurBlk, ca1w1, ca1w2, scale);
  k_cat<<<nPixBlk, 256, 0, stream>>>(hfb, tmp12, w_cat, b_cat, scale, f12);
  k_c9a<<<nPixBlk, 256, 0, stream>>>(f12, w_c9a, b_c9a, hbuf);
  k_c9b_wmma<<<32 * 32, 256, 0, stream>>>(hbuf, wp9, b_c9b, out);
}